// HashEncoder_11587821765207
// MI455X (gfx1250) — compile-verified
//
#include <hip/hip_runtime.h>
#include <stdint.h>

// ---------------------------------------------------------------------------
// InstantNGP hash-grid encoder for gfx1250 (MI455X).
//   16 levels, F=2, hashmap 2^19, base res 16, finest 512.
//   All table sizes are powers of two -> hash mod == AND mask, uint32 math
//   reproduces the reference's int64 hash exactly (only low <=19 bits used).
//   Levels 5..15 share res=512/mask 2^19-1 -> hash+weights computed once.
//   Levels 0+1 (36864 entries, 288KB) staged in LDS via async-to-LDS loads.
// ---------------------------------------------------------------------------

#define TPB 1024
#define L01_ENTRIES 36864        // 4096 (level0) + 32768 (level1)

typedef int v4i __attribute__((vector_size(16)));
typedef v4i __attribute__((address_space(1))) gv4i;   // global AS pointer elem
typedef v4i __attribute__((address_space(3))) lv4i;   // LDS AS pointer elem

__device__ __forceinline__ void hg_corners(float px, float py, float pz,
                                           int res, uint32_t mask,
                                           uint32_t h[8], float w[8])
{
    const float rs = (float)(res - 1);
    const float sx = px * rs, sy = py * rs, sz = pz * rs;
    const float fx = floorf(sx), fy = floorf(sy), fz = floorf(sz);
    const int ix = (int)fx, iy = (int)fy, iz = (int)fz;
    const float wx = sx - fx, wy = sy - fy, wz = sz - fz;
    const int r1 = res - 1;
    const int jx = min(ix + 1, r1), jy = min(iy + 1, r1), jz = min(iz + 1, r1);

    const uint32_t hx0 = (uint32_t)ix * 73856093u;
    const uint32_t hx1 = (uint32_t)jx * 73856093u;
    const uint32_t hy0 = (uint32_t)iy * 19349663u;
    const uint32_t hy1 = (uint32_t)jy * 19349663u;
    const uint32_t hz0 = (uint32_t)iz * 83492791u;
    const uint32_t hz1 = (uint32_t)jz * 83492791u;

    const float ux = 1.0f - wx, uy = 1.0f - wy, uz = 1.0f - wz;

    // Corner order: dx outer, dy, dz inner (matches reference CORNERS).
    h[0] = (hx0 + hy0 + hz0) & mask;  w[0] = ux * uy * uz;
    h[1] = (hx0 + hy0 + hz1) & mask;  w[1] = ux * uy * wz;
    h[2] = (hx0 + hy1 + hz0) & mask;  w[2] = ux * wy * uz;
    h[3] = (hx0 + hy1 + hz1) & mask;  w[3] = ux * wy * wz;
    h[4] = (hx1 + hy0 + hz0) & mask;  w[4] = wx * uy * uz;
    h[5] = (hx1 + hy0 + hz1) & mask;  w[5] = wx * uy * wz;
    h[6] = (hx1 + hy1 + hz0) & mask;  w[6] = wx * wy * uz;
    h[7] = (hx1 + hy1 + hz1) & mask;  w[7] = wx * wy * wz;
}

__device__ __forceinline__ float2 hg_gather_global(const float2* __restrict__ tab,
                                                   const uint32_t h[8],
                                                   const float w[8])
{
    float ax = 0.0f, ay = 0.0f;
#pragma unroll
    for (int c = 0; c < 8; ++c) {
        const float2 e = tab[h[c]];
        ax = fmaf(w[c], e.x, ax);
        ay = fmaf(w[c], e.y, ay);
    }
    return make_float2(ax, ay);
}

__global__ __launch_bounds__(TPB, 1)
void hashgrid_encode(const float* __restrict__ positions,
                     const float* __restrict__ embeddings,
                     const float* __restrict__ aabb_min,
                     const float* __restrict__ aabb_max,
                     float* __restrict__ out, int N)
{
    __shared__ float2 s_tab[L01_ENTRIES];   // 288 KB: levels 0 and 1

    const int tid = threadIdx.x;

    // ---- Stage levels 0+1 into LDS (async tensor path, ASYNCcnt) ----------
    {
        float* ldsf = (float*)s_tab;
#if __has_builtin(__builtin_amdgcn_global_load_async_to_lds_b128)
#pragma unroll
        for (int i = 0; i < (L01_ENTRIES * 2) / 4 / TPB; ++i) {   // 18 iters
            const int idx4 = (i * TPB + tid) * 4;
            __builtin_amdgcn_global_load_async_to_lds_b128(
                (gv4i*)(embeddings + idx4),
                (lv4i*)(ldsf + idx4),
                0, 0);
        }
#if __has_builtin(__builtin_amdgcn_s_wait_asynccnt)
        __builtin_amdgcn_s_wait_asynccnt(0);
#else
        asm volatile("s_wait_asynccnt 0" ::: "memory");
#endif
#else
        // Fallback: synchronous staging through VGPRs.
        const float4* src = (const float4*)embeddings;
        float4* dst = (float4*)ldsf;
        for (int i = tid; i < (L01_ENTRIES * 2) / 4; i += TPB)
            dst[i] = src[i];
#endif
    }
    __syncthreads();

    const int n = blockIdx.x * TPB + tid;
    if (n >= N) return;

    // ---- Normalize position into [0,1]^3 (matches jnp.clip((p-mn)/(mx-mn))) --
    const float mn0 = aabb_min[0], mn1 = aabb_min[1], mn2 = aabb_min[2];
    const float mx0 = aabb_max[0], mx1 = aabb_max[1], mx2 = aabb_max[2];
    float px = positions[3 * (size_t)n + 0];
    float py = positions[3 * (size_t)n + 1];
    float pz = positions[3 * (size_t)n + 2];
    px = fminf(fmaxf((px - mn0) / (mx0 - mn0), 0.0f), 1.0f);
    py = fminf(fmaxf((py - mn1) / (mx1 - mn1), 0.0f), 1.0f);
    pz = fminf(fmaxf((pz - mn2) / (mx2 - mn2), 0.0f), 1.0f);

    const float2* etab = (const float2*)embeddings;

    float feats[32];
    uint32_t h[8];
    float w[8];

    // ---- Level 0 (res 16, LDS) --------------------------------------------
    hg_corners(px, py, pz, 16, 4095u, h, w);
    {
        float ax = 0.0f, ay = 0.0f;
#pragma unroll
        for (int c = 0; c < 8; ++c) {
            const float2 e = s_tab[h[c]];          // ds_load_b64
            ax = fmaf(w[c], e.x, ax);
            ay = fmaf(w[c], e.y, ay);
        }
        feats[0] = ax; feats[1] = ay;
    }

    // ---- Level 1 (res 32, LDS, entry offset 4096) -------------------------
    hg_corners(px, py, pz, 32, 32767u, h, w);
    {
        float ax = 0.0f, ay = 0.0f;
#pragma unroll
        for (int c = 0; c < 8; ++c) {
            const float2 e = s_tab[4096u + h[c]];  // ds_load_b64
            ax = fmaf(w[c], e.x, ax);
            ay = fmaf(w[c], e.y, ay);
        }
        feats[2] = ax; feats[3] = ay;
    }

    // ---- Levels 2..4 (global, per-level hash) -----------------------------
    hg_corners(px, py, pz, 64, 262143u, h, w);
    { float2 f = hg_gather_global(etab + 36864u,  h, w); feats[4] = f.x; feats[5] = f.y; }

    hg_corners(px, py, pz, 128, 524287u, h, w);
    { float2 f = hg_gather_global(etab + 299008u, h, w); feats[6] = f.x; feats[7] = f.y; }

    hg_corners(px, py, pz, 256, 524287u, h, w);
    { float2 f = hg_gather_global(etab + 823296u, h, w); feats[8] = f.x; feats[9] = f.y; }

    // ---- Levels 5..15: identical res=512 geometry, hash computed once -----
    hg_corners(px, py, pz, 512, 524287u, h, w);
#pragma unroll
    for (int l = 5; l < 16; ++l) {
        const float2* tab = etab + 1347584u + (uint32_t)(l - 5) * 524288u;
        const float2 f = hg_gather_global(tab, h, w);
        feats[2 * l]     = f.x;
        feats[2 * l + 1] = f.y;
    }

    // ---- Write [n, 32] as 8 x b128 stores ---------------------------------
    float4* o = (float4*)(out + (size_t)n * 32);
#pragma unroll
    for (int i = 0; i < 8; ++i)
        o[i] = make_float4(feats[4 * i + 0], feats[4 * i + 1],
                           feats[4 * i + 2], feats[4 * i + 3]);
}

extern "C" void kernel_launch(void* const* d_in, const int* in_sizes, int n_in,
                              void* d_out, int out_size, void* d_ws, size_t ws_size,
                              hipStream_t stream)
{
    const float* positions  = (const float*)d_in[0];
    const float* embeddings = (const float*)d_in[1];
    const float* amin       = (const float*)d_in[2];
    const float* amax       = (const float*)d_in[3];
    const int N = in_sizes[0] / 3;

    const int blocks = (N + TPB - 1) / TPB;
    hipLaunchKernelGGL(hashgrid_encode, dim3(blocks), dim3(TPB), 0, stream,
                       positions, embeddings, amin, amax, (float*)d_out, N);
}